// ChunkCausalDepthwiseConv1d_2267742733074
// MI455X (gfx1250) — compile-verified
//
#include <hip/hip_runtime.h>
#include <stdint.h>

typedef __attribute__((ext_vector_type(2)))  float        v2f;
typedef __attribute__((ext_vector_type(8)))  float        v8f;
typedef __attribute__((ext_vector_type(4)))  unsigned int u32x4;
typedef __attribute__((ext_vector_type(4)))  int          i32x4;
typedef __attribute__((ext_vector_type(8)))  int          i32x8;

#define T_LEN   4096
#define C_CH    512
#define NB      16
#define KT      31      // chunk conv taps
#define HK      16      // causal conv taps
#define CHUNK   256
#define SEG     2048    // time-steps per workgroup (8 waves x 1 chunk)
#define HALO    15
#define TILE_F  2080    // halo'd tile for causal conv: x[seg-15 .. seg+2065)
#define PCHUNK  288     // padded chunk stride: 15 zeros + 256 data + 17 zeros
#define PT_F    2320    // 15 + 8*288 = 2319, rounded up

typedef const __attribute__((address_space(3))) void* lds_cvptr;

__device__ __forceinline__ unsigned lds_byte_off(const void* p) {
  return (unsigned)(uintptr_t)(lds_cvptr)p;
}

__global__ __launch_bounds__(256)
void chunk_causal_dwconv_kernel(const float* __restrict__ x,
                                const float* __restrict__ causal_w,
                                const float* __restrict__ causal_b,
                                const float* __restrict__ chunk_w,
                                const float* __restrict__ chunk_b,
                                const float* __restrict__ conv_scale,
                                float* __restrict__ out) {
  __shared__ __align__(16) float tile[TILE_F];   // halo'd x (causal conv B)
  __shared__ __align__(16) float ptile[PT_F];    // chunk-padded x (chunk conv B)
  __shared__ __align__(16) float stage[SEG];     // output transpose staging
  __shared__ float wext[64];                     // chunk weights, zero-extended
  __shared__ float cext[48];                     // causal weights, zero-extended
  __shared__ float scl[CHUNK];                   // per-position chunk scale

  const int tid  = threadIdx.x;
  const int flat = blockIdx.x;
  const int segi = flat & 1;                 // which half of the row
  const int row  = flat >> 1;                // b*C + c
  const int c    = row & (C_CH - 1);
  const int seg  = segi * SEG;
  const long rowbase = (long)row * T_LEN;

  // ---- precompute per-position chunk scale (all 256 threads) ----
  {
    float s = 1.0f;
    if (tid < KT)          s += conv_scale[c * KT + tid];
    if (tid >= CHUNK - KT) s += conv_scale[(C_CH + c) * KT + (tid - (CHUNK - KT))];
    scl[tid] = s;
  }

  // ---- zero-extended weight vectors: A[m,j] = wext[15 + j - m], unmasked ----
  if (tid < 64) {
    const int d = tid - HALO;
    wext[tid] = (d >= 0 && d < KT) ? chunk_w[c * KT + d] : 0.0f;
  } else if (tid < 112) {
    const int i = tid - 64;
    const int d = i - HALO;
    cext[i] = (d >= 0 && d < HK) ? causal_w[c * HK + d] : 0.0f;
  }

  // ---- zero the padding aprons of ptile (15 head + 32 per inter-chunk gap) ----
  for (int i = tid; i < HALO + 8 * 32; i += 256) {
    int idx;
    if (i < HALO) idx = i;
    else {
      const int j = i - HALO;
      idx = (j >> 5) * PCHUNK + (HALO + CHUNK) + (j & 31);   // [ch*288+271, ch*288+303)
    }
    ptile[idx] = 0.0f;
  }

  // zero the left halo of the plain tile for the first segment (t < 0)
  if (segi == 0 && tid < HALO) tile[tid] = 0.0f;

  // ---- TDM: two async tensor loads into LDS, wave 0 only ----
  if (tid < 32) {
    // (a) halo'd tile for the causal conv, OOB zero-fill past row end
    {
      const int      gstart = (segi == 0) ? 0 : (seg - HALO);
      const int      ldsoff = (segi == 0) ? HALO : 0;
      const unsigned tl0    = (unsigned)(TILE_F - ldsoff);
      const unsigned td0    = (unsigned)(T_LEN - gstart);
      const uint64_t ga     = (uint64_t)(uintptr_t)(x + rowbase + gstart);
      u32x4 g0;
      g0[0] = 1u;                                          // count = 1
      g0[1] = lds_byte_off(tile) + (unsigned)ldsoff * 4u;  // lds_addr (bytes)
      g0[2] = (unsigned)(ga & 0xffffffffull);
      g0[3] = (unsigned)((ga >> 32) & 0x01ffffffull) | 0x80000000u;  // type=2
      i32x8 g1;
      g1[0] = 0x00020000;                       // data_size = 4B
      g1[1] = (int)(td0 << 16);                 // tensor_dim0[15:0]
      g1[2] = (int)((td0 >> 16) | (1u << 16));  // tensor_dim0 hi | tensor_dim1 = 1
      g1[3] = (int)(tl0 << 16);                 // tile_dim0
      g1[4] = 0;
      g1[5] = (int)td0;                         // tensor_dim0_stride
      g1[6] = 0;  g1[7] = 0;
      i32x4 z4 = {0, 0, 0, 0};
#if __clang_major__ >= 23
      i32x8 z8 = {0, 0, 0, 0, 0, 0, 0, 0};
      __builtin_amdgcn_tensor_load_to_lds(g0, g1, z4, z4, z8, 0);
#else
      __builtin_amdgcn_tensor_load_to_lds(g0, g1, z4, z4, 0);
#endif
    }
    // (b) chunk-padded tile: LDS padding inserts the 32-dword apron per chunk
    {
      const unsigned td0 = (unsigned)SEG;                   // exactly in-bounds
      const uint64_t ga  = (uint64_t)(uintptr_t)(x + rowbase + seg);
      u32x4 g0;
      g0[0] = 1u;
      g0[1] = lds_byte_off(ptile) + HALO * 4u;              // data starts after 15 zeros
      g0[2] = (unsigned)(ga & 0xffffffffull);
      g0[3] = (unsigned)((ga >> 32) & 0x01ffffffull) | 0x80000000u;
      i32x8 g1;
      // data_size=4B | pad_enable | pad_interval=7 (256 dw) | pad_amount=31 (32 dw)
      g1[0] = (int)0x3FD20000u;
      g1[1] = (int)(td0 << 16);
      g1[2] = (int)((td0 >> 16) | (1u << 16));
      g1[3] = (int)(td0 << 16);                             // tile_dim0 = 2048
      g1[4] = 0;
      g1[5] = (int)td0;
      g1[6] = 0;  g1[7] = 0;
      i32x4 z4 = {0, 0, 0, 0};
#if __clang_major__ >= 23
      i32x8 z8 = {0, 0, 0, 0, 0, 0, 0, 0};
      __builtin_amdgcn_tensor_load_to_lds(g0, g1, z4, z4, z8, 0);
#else
      __builtin_amdgcn_tensor_load_to_lds(g0, g1, z4, z4, 0);
#endif
    }
    __builtin_amdgcn_s_wait_tensorcnt(0);
  }
  __syncthreads();

  // ---- per-wave WMMA conv: wave wv owns chunk wv of this segment ----
  const int wv   = tid >> 5;
  const int lane = tid & 31;
  const int hh   = lane >> 4;       // half-wave (selects K pair / M+8)
  const int nn   = lane & 15;       // A: row m; B: col n
  const int t0l  = wv * CHUNK;      // chunk start inside segment (plain tile)
  const int pb   = wv * PCHUNK;     // chunk base inside padded tile

  // chunk conv: D[m,n] = sum_j wext[15+j-m] * ptile[pb + n*16 + j], j in [0,48)
  v8f dc = {};
#pragma unroll
  for (int g = 0; g < 12; ++g) {
    const int jb = 4 * g + 2 * hh;
    v2f a, bv;
#pragma unroll
    for (int r = 0; r < 2; ++r) {
      const int j = jb + r;
      a[r]  = wext[HALO + j - nn];
      bv[r] = ptile[pb + nn * 16 + j];
    }
    dc = __builtin_amdgcn_wmma_f32_16x16x4_f32(false, a, false, bv,
                                               (short)0, dc, false, false);
  }

  // causal conv: D[m,n] = sum_j cext[15+j-m] * tile[t0l + n*16 + j], j in [0,32)
  v8f dk = {};
#pragma unroll
  for (int g = 0; g < 8; ++g) {
    const int jb = 4 * g + 2 * hh;
    v2f a, bv;
#pragma unroll
    for (int r = 0; r < 2; ++r) {
      const int j = jb + r;
      a[r]  = cext[HALO + j - nn];
      bv[r] = tile[t0l + nn * 16 + j];
    }
    dk = __builtin_amdgcn_wmma_f32_16x16x4_f32(false, a, false, bv,
                                               (short)0, dk, false, false);
  }

  // ---- epilogue: bias + scale + combine, transpose via LDS, coalesced store ----
  const float cbv = chunk_b[c];
  const float kbv = causal_b[c];
#pragma unroll
  for (int r = 0; r < 8; ++r) {
    const int m = r + 8 * hh;          // C/D layout: VGPR r, lanes16-31 -> M+8
    const int p = nn * 16 + m;         // position within chunk [0,256)
    stage[t0l + p] = (dc[r] + cbv) * scl[p] + dk[r] + kbv;
  }

  // stage region is wave-private; same-wave LDS ops are in order (DScnt)
#pragma unroll
  for (int i = 0; i < 2; ++i) {
    const int off = i * 128 + lane * 4;
    const float4 v = *(const float4*)&stage[t0l + off];
    *(float4*)&out[rowbase + seg + t0l + off] = v;
  }
}

extern "C" void kernel_launch(void* const* d_in, const int* in_sizes, int n_in,
                              void* d_out, int out_size, void* d_ws, size_t ws_size,
                              hipStream_t stream) {
  const float* x  = (const float*)d_in[0];
  const float* cw = (const float*)d_in[1];
  const float* cb = (const float*)d_in[2];
  const float* hw = (const float*)d_in[3];
  const float* hb = (const float*)d_in[4];
  const float* cs = (const float*)d_in[5];
  float* out = (float*)d_out;

  const int nblocks = NB * C_CH * (T_LEN / SEG);   // 16384 workgroups
  chunk_causal_dwconv_kernel<<<dim3(nblocks), dim3(256), 0, stream>>>(
      x, cw, cb, hw, hb, cs, out);
}